// STSGT_28114855919850
// MI455X (gfx1250) — compile-verified
//
#include <hip/hip_runtime.h>
#include <hip/hip_bf16.h>

// ---------------- problem constants ----------------
#define NN 10000
#define TT 8
#define CC 64
#define HH 4
#define DD 16
#define EE 160000
#define NT (NN*TT)                    // 80000 rows
#define NTCf ((size_t)NT*CC)          // 5,120,000 floats
#define NTHf ((size_t)NT*HH)          // 320,000 floats

typedef float v2f __attribute__((ext_vector_type(2)));
typedef float v8f __attribute__((ext_vector_type(8)));

// ---------------- fp32 WMMA 16x16 tile ----------------
// A: 16 x K in LDS, row-major with padded pitch lda (bank-conflict-free when
// lda % 64 == 4).  B: stored TRANSPOSED in LDS as [ncols][K] with padded pitch
// ldbt; sBt points at the 16-column tile base.  Each lane then feeds both the
// A pair and the B pair with a single ds_load_b64.
// Layouts per CDNA5 ISA 7.12.2 (32-bit A 16x4, B 4x16, C/D 16x16).
template<int K>
__device__ __forceinline__ v8f wmma_tile_T(const float* __restrict__ sA, int lda,
                                           const float* __restrict__ sBt, int ldbt,
                                           int lane)
{
    const int m  = lane & 15;                 // A row / D col
    const int n  = lane & 15;                 // B col
    const int kh = (lane & 16) ? 2 : 0;       // K-pair per half-wave
    const float* arow = sA  + m*lda  + kh;
    const float* brow = sBt + n*ldbt + kh;
    v8f acc = {0.f,0.f,0.f,0.f,0.f,0.f,0.f,0.f};
#pragma unroll
    for (int k0 = 0; k0 < K; k0 += 4) {
        v2f a = *(const v2f*)(arow + k0);     // A[m][k0+kh], A[m][k0+kh+1]
        v2f b = *(const v2f*)(brow + k0);     // B[k0+kh][n], B[k0+kh+1][n]
        acc = __builtin_amdgcn_wmma_f32_16x16x4_f32(false, a, false, b,
                                                    (short)0, acc, false, false);
    }
    return acc;
}

// ---- async global->LDS staging of a 16x64 f32 tile into a padded [16][68] LDS
// buffer, using CDNA5 GLOBAL_LOAD_ASYNC_TO_LDS_B128 (tracked by ASYNCcnt).
// 256 16-byte chunks; 128 threads x 2 chunks. LDS byte address = low 32 bits
// of the flat shared pointer (LDS aperture maps offset into addr[31:0]).
__device__ __forceinline__ void async_tile16x64(float* sdst, const float* gsrc, int tid)
{
#pragma unroll
    for (int c = tid; c < 256; c += 128) {
        unsigned lds = (unsigned)(uintptr_t)(sdst + (c >> 4)*68 + (c & 15)*4);
        const float* g = gsrc + (size_t)c*4;
        asm volatile("global_load_async_to_lds_b128 %0, %1, off"
                     :: "v"(lds), "v"(g) : "memory");
    }
}
__device__ __forceinline__ void async_wait_all()
{
    asm volatile("s_wait_asynccnt 0x0" ::: "memory");
}

__device__ __forceinline__ float dot4(float4 a, float4 b){
    return a.x*b.x + a.y*b.y + a.z*b.z + a.w*b.w;
}
__device__ __forceinline__ void fma4(float4& o, float w, float4 v){
    o.x += w*v.x; o.y += w*v.y; o.z += w*v.z; o.w += w*v.w;
}

// order-preserving float<->uint map for atomic segment-max
__device__ __forceinline__ unsigned fmap(float f){
    unsigned u = __float_as_uint(f);
    return (u & 0x80000000u) ? ~u : (u | 0x80000000u);
}
__device__ __forceinline__ float funmap(unsigned u){
    unsigned b = (u & 0x80000000u) ? (u & 0x7FFFFFFFu) : ~u;
    return __uint_as_float(b);
}

// ---------------- 1) init projection: h[n,t,c] = x[n,t]*w[c] + b[c] ----------------
__global__ void __launch_bounds__(256)
k_init(const float* __restrict__ x, const float* __restrict__ iw,
       const float* __restrict__ ib, float* __restrict__ h)
{
    size_t idx = (size_t)blockIdx.x*256 + threadIdx.x;
    if (idx >= NTCf) return;
    int c = (int)(idx & 63);
    size_t nt = idx >> 6;                 // flattened (n*T + t)
    h[idx] = x[nt]*iw[c] + ib[c];
}

// ---------------- 2) WMMA GEMM: out[NTxNC] = A[NTx64] @ W[64xNC] + bias ------------
// W staged transposed+padded in LDS; A tile staged via async global->LDS.
template<int NC>
__global__ void __launch_bounds__(128)
k_gemm_bias(const float* __restrict__ A, const float* __restrict__ W,
            const float* __restrict__ bias, float* __restrict__ out)
{
    __shared__ __align__(16) float sWt[NC*68];     // [NC][68]: sWt[n][k] = W[k][n]
    __shared__ __align__(16) float sA[16*68];      // [16][68]
    const int tid = threadIdx.x;
    const size_t r0 = (size_t)blockIdx.x * 16;
    async_tile16x64(sA, A + r0*64, tid);           // ASYNCcnt-tracked DMA
    for (int i = tid; i < 64*NC; i += 128) {       // transpose W into LDS
        int k = i / NC, n = i - k*NC;
        sWt[n*68 + k] = W[i];
    }
    async_wait_all();
    __syncthreads();
    const int wave = tid >> 5, lane = tid & 31;
    const int n = lane & 15, hi = (lane & 16) ? 8 : 0;
#pragma unroll
    for (int tc = wave; tc < (NC >> 4); tc += 4) {
        v8f acc = wmma_tile_T<64>(sA, 68, sWt + tc*16*68, 68, lane);
        const int col = tc*16 + n;
        const float bv = bias[col];
#pragma unroll
        for (int i = 0; i < 8; i++)
            out[(r0 + i + hi)*NC + col] = acc[i] + bv;
    }
}

// ---------------- 3) temporal MHSA per (node, head, t): 8x8 softmax attention ------
__global__ void __launch_bounds__(256)
k_temporal_attn(const float* __restrict__ qkv, float* __restrict__ O)
{
    const int tid  = threadIdx.x;
    const int node = blockIdx.x*8 + (tid >> 5);
    const int sub  = tid & 31;
    const int hh   = sub >> 3;            // head 0..3
    const int t    = sub & 7;             // query timestep
    const float* base = qkv + (size_t)node*TT*192;
    const float4* qp = (const float4*)(base + t*192 + hh*16);
    float4 q0 = qp[0], q1 = qp[1], q2 = qp[2], q3 = qp[3];
    float sc[TT]; float mx = -3.4e38f;
#pragma unroll
    for (int s = 0; s < TT; s++) {
        const float4* kp = (const float4*)(base + s*192 + 64 + hh*16);
        float d = dot4(q0,kp[0]) + dot4(q1,kp[1]) + dot4(q2,kp[2]) + dot4(q3,kp[3]);
        sc[s] = d * 0.25f;                // 1/sqrt(16)
        mx = fmaxf(mx, sc[s]);
    }
    float sum = 0.f;
#pragma unroll
    for (int s = 0; s < TT; s++) { sc[s] = __expf(sc[s]-mx); sum += sc[s]; }
    const float inv = 1.f/sum;
    float4 o0={0,0,0,0}, o1={0,0,0,0}, o2={0,0,0,0}, o3={0,0,0,0};
#pragma unroll
    for (int s = 0; s < TT; s++) {
        const float4* vp = (const float4*)(base + s*192 + 128 + hh*16);
        float w = sc[s]*inv;
        fma4(o0, w, vp[0]); fma4(o1, w, vp[1]); fma4(o2, w, vp[2]); fma4(o3, w, vp[3]);
    }
    float4* op = (float4*)(O + ((size_t)node*TT + t)*64 + hh*16);
    op[0]=o0; op[1]=o1; op[2]=o2; op[3]=o3;
}

// ---------------- 4) fused o@Wout + bias + residual + LayerNorm (in-place h) -------
__global__ void __launch_bounds__(128)
k_attnout_ln(const float* __restrict__ O, const float* __restrict__ W,
             const float* __restrict__ bias, const float* __restrict__ g,
             const float* __restrict__ be, float* __restrict__ Hio)
{
    __shared__ __align__(16) float sWt[64*68];     // transposed Wout
    __shared__ __align__(16) float sA[16*68];      // o rows (padded)
    __shared__ __align__(16) float sH[16*68];      // h rows (padded)
    __shared__ __align__(16) float sP[16*64];      // projected tile
    const int tid = threadIdx.x;
    const size_t r0 = (size_t)blockIdx.x*16;
    async_tile16x64(sA, O   + r0*64, tid);
    async_tile16x64(sH, Hio + r0*64, tid);
    for (int i = tid; i < 64*64; i += 128) {
        int k = i >> 6, n = i & 63;
        sWt[n*68 + k] = W[i];
    }
    async_wait_all();
    __syncthreads();
    const int wave = tid >> 5, lane = tid & 31;
    {
        v8f acc = wmma_tile_T<64>(sA, 68, sWt + wave*16*68, 68, lane);
        const int n = lane & 15, col = wave*16 + n, hi = (lane & 16) ? 8 : 0;
        const float bv = bias[col];
#pragma unroll
        for (int i = 0; i < 8; i++) sP[(i+hi)*64 + col] = acc[i] + bv;
    }
    __syncthreads();
    if (tid < 16) {
        const int row = tid;
        float mu = 0.f;
        for (int c = 0; c < 64; c++) { float v = sH[row*68+c] + sP[row*64+c]; sP[row*64+c] = v; mu += v; }
        mu *= (1.f/64.f);
        float var = 0.f;
        for (int c = 0; c < 64; c++) { float d = sP[row*64+c]-mu; var += d*d; }
        var *= (1.f/64.f);
        const float rs = rsqrtf(var + 1e-5f);
        for (int c = 0; c < 64; c++)
            Hio[(r0+row)*64 + c] = (sP[row*64+c]-mu)*rs*g[c] + be[c];
    }
}

// ---------------- 5) edge phase: logits + segment max ------------------------------
__global__ void __launch_bounds__(256)
k_edge_logit_max(const int* __restrict__ ei, const float* __restrict__ q,
                 const float* __restrict__ k, float* __restrict__ lg,
                 unsigned* __restrict__ mb)
{
    size_t gid = (size_t)blockIdx.x*256 + threadIdx.x;
    if (gid >= (size_t)EE*32) return;
    const int e = (int)(gid >> 5), sub = (int)(gid & 31);
    const int t = sub >> 2, hh = sub & 3;
    const int src = ei[e], dst = ei[EE + e];
    const float4* qp = (const float4*)(q + ((size_t)(dst*TT + t))*64 + hh*16);
    const float4* kp = (const float4*)(k + ((size_t)(src*TT + t))*64 + hh*16);
    float d = dot4(qp[0],kp[0]) + dot4(qp[1],kp[1]) + dot4(qp[2],kp[2]) + dot4(qp[3],kp[3]);
    float logit = d*0.25f;
    lg[gid] = logit;
    atomicMax(&mb[(dst*TT + t)*HH + hh], fmap(logit));
}

// ---------------- 6) edge phase: exp + segment sum ---------------------------------
__global__ void __launch_bounds__(256)
k_edge_ex_den(const int* __restrict__ ei, float* __restrict__ lg,
              const unsigned* __restrict__ mb, float* __restrict__ den)
{
    size_t gid = (size_t)blockIdx.x*256 + threadIdx.x;
    if (gid >= (size_t)EE*32) return;
    const int e = (int)(gid >> 5), sub = (int)(gid & 31);
    const int t = sub >> 2, hh = sub & 3;
    const int dst = ei[EE + e];
    const int ms = (dst*TT + t)*HH + hh;
    float ex = __expf(lg[gid] - funmap(mb[ms]));
    lg[gid] = ex;
    atomicAdd(&den[ms], ex);
}

// ---------------- 7) edge phase: alpha * v[src] scatter-add ------------------------
__global__ void __launch_bounds__(256)
k_edge_agg(const int* __restrict__ ei, const float* __restrict__ lg,
           const float* __restrict__ den, const float* __restrict__ v,
           float* __restrict__ agg)
{
    size_t gid = (size_t)blockIdx.x*256 + threadIdx.x;
    if (gid >= (size_t)EE*32) return;
    const int e = (int)(gid >> 5), sub = (int)(gid & 31);
    const int t = sub >> 2, hh = sub & 3;
    const int src = ei[e], dst = ei[EE + e];
    const int ms = (dst*TT + t)*HH + hh;
    const float alpha = lg[gid] / (den[ms] + 1e-16f);
    const float* vp = v + ((size_t)(src*TT + t))*64 + hh*16;
    float* ap = agg + ((size_t)(dst*TT + t))*64 + hh*16;
#pragma unroll
    for (int d2 = 0; d2 < 16; d2++)
        atomicAdd(&ap[d2], alpha * vp[d2]);
}

// ---------------- 8) h = LN(h + agg + skip) (in-place) -----------------------------
__global__ void __launch_bounds__(128)
k_add_ln(const float* __restrict__ Agg, const float* __restrict__ Skip,
         const float* __restrict__ g, const float* __restrict__ be,
         float* __restrict__ Hio)
{
    __shared__ float sV[16*64];
    const int tid = threadIdx.x;
    const size_t r0 = (size_t)blockIdx.x*16;
    for (int i = tid; i < 1024; i += 128) {
        size_t gi = r0*64 + i;
        sV[i] = Hio[gi] + Agg[gi] + Skip[gi];
    }
    __syncthreads();
    if (tid < 16) {
        const int row = tid;
        float mu = 0.f;
        for (int c = 0; c < 64; c++) mu += sV[row*64+c];
        mu *= (1.f/64.f);
        float var = 0.f;
        for (int c = 0; c < 64; c++) { float d = sV[row*64+c]-mu; var += d*d; }
        var *= (1.f/64.f);
        const float rs = rsqrtf(var + 1e-5f);
        for (int c = 0; c < 64; c++)
            Hio[(r0+row)*64 + c] = (sV[row*64+c]-mu)*rs*g[c] + be[c];
    }
}

// ---------------- 9) fused FFN: h += relu(h@W1+b1)@W2+b2 (in-place) ----------------
__global__ void __launch_bounds__(128)
k_ffn(const float* __restrict__ W1, const float* __restrict__ b1,
      const float* __restrict__ W2, const float* __restrict__ b2,
      float* __restrict__ Hio)
{
    __shared__ __align__(16) float sW1t[128*68];   // [128][68]: W1 transposed, K=64
    __shared__ __align__(16) float sW2t[64*132];   // [64][132]: W2 transposed, K=128
    __shared__ __align__(16) float sA[16*68];      // h rows (padded)
    __shared__ __align__(16) float sM[16*132];     // relu mid tile (padded, K=128)
    const int tid = threadIdx.x;
    const size_t r0 = (size_t)blockIdx.x*16;
    async_tile16x64(sA, Hio + r0*64, tid);
    for (int i = tid; i < 64*128; i += 128) {      // W1: [64][128] -> sW1t[n][k]
        int k = i >> 7, n = i & 127;
        sW1t[n*68 + k] = W1[i];
    }
    for (int i = tid; i < 128*64; i += 128) {      // W2: [128][64] -> sW2t[n][k]
        int k = i >> 6, n = i & 63;
        sW2t[n*132 + k] = W2[i];
    }
    async_wait_all();
    __syncthreads();
    const int wave = tid >> 5, lane = tid & 31;
    const int n = lane & 15, hi = (lane & 16) ? 8 : 0;
#pragma unroll
    for (int tc = wave; tc < 8; tc += 4) {         // mid: 16 x 128, relu
        v8f acc = wmma_tile_T<64>(sA, 68, sW1t + tc*16*68, 68, lane);
        const int col = tc*16 + n; const float bv = b1[col];
#pragma unroll
        for (int i = 0; i < 8; i++) sM[(i+hi)*132 + col] = fmaxf(acc[i] + bv, 0.f);
    }
    __syncthreads();
    {                                              // out: 16 x 64, K=128
        v8f acc = wmma_tile_T<128>(sM, 132, sW2t + wave*16*132, 132, lane);
        const int col = wave*16 + n; const float bv = b2[col];
#pragma unroll
        for (int i = 0; i < 8; i++) {
            const int rl = i + hi;
            Hio[(r0+rl)*64 + col] = acc[i] + bv + sA[rl*68 + col];
        }
    }
}

// ---------------- 10) regression head on last timestep -----------------------------
__global__ void __launch_bounds__(256)
k_head(const float* __restrict__ H, const float* __restrict__ lw,
       const float* __restrict__ lb, float* __restrict__ out)
{
    const int tid = threadIdx.x, lane = tid & 31;
    const int node = blockIdx.x*8 + (tid >> 5);
    const float* row = H + ((size_t)node*TT + (TT-1))*64;
    float s = row[lane]*lw[lane] + row[lane+32]*lw[lane+32];
#pragma unroll
    for (int off = 16; off > 0; off >>= 1) s += __shfl_down(s, off, 32);
    if (lane == 0) out[node] = s + lb[0];
}

// ---------------- launcher ----------------------------------------------------------
extern "C" void kernel_launch(void* const* d_in, const int* in_sizes, int n_in,
                              void* d_out, int out_size, void* d_ws, size_t ws_size,
                              hipStream_t stream)
{
    (void)in_sizes; (void)n_in; (void)out_size; (void)ws_size;
    const float* x          = (const float*)d_in[0];
    const int*   ei         = (const int*)  d_in[1];
    // d_in[2] edge_weight: unused by the reference forward
    const float* init_w     = (const float*)d_in[3];
    const float* init_b     = (const float*)d_in[4];
    const float* attn_in_w  = (const float*)d_in[5];
    const float* attn_in_b  = (const float*)d_in[6];
    const float* attn_out_w = (const float*)d_in[7];
    const float* attn_out_b = (const float*)d_in[8];
    const float* n1_g       = (const float*)d_in[9];
    const float* n1_b       = (const float*)d_in[10];
    const float* wq         = (const float*)d_in[11];
    const float* bq         = (const float*)d_in[12];
    const float* wk         = (const float*)d_in[13];
    const float* bk         = (const float*)d_in[14];
    const float* wv         = (const float*)d_in[15];
    const float* bv         = (const float*)d_in[16];
    const float* wskip      = (const float*)d_in[17];
    const float* bskip      = (const float*)d_in[18];
    const float* n2_g       = (const float*)d_in[19];
    const float* n2_b       = (const float*)d_in[20];
    const float* ffn_w1     = (const float*)d_in[21];
    const float* ffn_b1     = (const float*)d_in[22];
    const float* ffn_w2     = (const float*)d_in[23];
    const float* ffn_b2     = (const float*)d_in[24];
    const float* lin_w      = (const float*)d_in[25];
    const float* lin_b      = (const float*)d_in[26];

    // workspace layout (floats), ~146 MB total, buffers reused across phases
    float* ws   = (float*)d_ws;
    float* h    = ws;                     // [NT,64]  h0 -> h1 -> h2 -> h3 (in-place)
    float* qkv  = ws + 1*NTCf;            // [NT,192] interleaved; later re-used as q
    float* qb   = ws + 1*NTCf;            // [NT,64]
    float* kb   = ws + 2*NTCf;            // [NT,64]
    float* vb   = ws + 3*NTCf;            // [NT,64]
    float* ob   = ws + 4*NTCf;            // [NT,64]  attn o; later skip
    float* lg   = ws + 5*NTCf;            // [E,T,H]  logits -> ex   (E*T*H == NT*C)
    float* agg  = ws + 6*NTCf;            // [NT,64]
    unsigned* mb= (unsigned*)(ws + 7*NTCf);      // [NT,H] mapped max
    float* den  = ws + 7*NTCf + NTHf;            // [NT,H]

    // zero agg + m + den (contiguous region)
    hipMemsetAsync(agg, 0, (NTCf + 2*NTHf)*sizeof(float), stream);

    k_init          <<<20000, 256, 0, stream>>>(x, init_w, init_b, h);
    k_gemm_bias<192><<< 5000, 128, 0, stream>>>(h, attn_in_w, attn_in_b, qkv);
    k_temporal_attn <<< 1250, 256, 0, stream>>>(qkv, ob);
    k_attnout_ln    <<< 5000, 128, 0, stream>>>(ob, attn_out_w, attn_out_b, n1_g, n1_b, h);
    k_gemm_bias<64> <<< 5000, 128, 0, stream>>>(h, wq,    bq,    qb);
    k_gemm_bias<64> <<< 5000, 128, 0, stream>>>(h, wk,    bk,    kb);
    k_gemm_bias<64> <<< 5000, 128, 0, stream>>>(h, wv,    bv,    vb);
    k_gemm_bias<64> <<< 5000, 128, 0, stream>>>(h, wskip, bskip, ob);   // skip
    k_edge_logit_max<<<20000, 256, 0, stream>>>(ei, qb, kb, lg, mb);
    k_edge_ex_den   <<<20000, 256, 0, stream>>>(ei, lg, mb, den);
    k_edge_agg      <<<20000, 256, 0, stream>>>(ei, lg, den, vb, agg);
    k_add_ln        <<< 5000, 128, 0, stream>>>(agg, ob, n2_g, n2_b, h);
    k_ffn           <<< 5000, 128, 0, stream>>>(ffn_w1, ffn_b1, ffn_w2, ffn_b2, h);
    k_head          <<< 1250, 256, 0, stream>>>(h, lin_w, lin_b, (float*)d_out);
}